// Detect_Flip_77653008711791
// MI455X (gfx1250) — compile-verified
//
#include <hip/hip_runtime.h>
#include <stdint.h>

#define BB 64
#define NN 8732
#define CC 21
#define N2 17464          // 2*NN
#define N2V4 4366         // N2/4 (exact)
#define TOPK 200
#define CONF_TH 0.01f
#define NMS_TH 0.45f
#define CHUNK 69          // ceil(N2/256)

__device__ __forceinline__ uint32_t fkey(float f) {
  // monotonic transform: float order == unsigned order
  uint32_t u = __float_as_uint(f);
  return (u & 0x80000000u) ? ~u : (u | 0x80000000u);
}
__device__ __forceinline__ float unkey(uint32_t k) {
  uint32_t u = (k & 0x80000000u) ? (k & 0x7FFFFFFFu) : ~k;
  return __uint_as_float(u);
}

// CDNA5 async data mover: per-lane 16B copy global -> LDS, tracked by ASYNCcnt.
// LDS address = low 32 bits of the generic shared pointer (ISA: generic LDS
// addr truncates to the 32-bit LDS offset).
__device__ __forceinline__ void async_copy_b128(void* lds_dst, const void* gsrc) {
  asm volatile("global_load_async_to_lds_b128 %0, %1, off"
               :: "v"((uint32_t)(uintptr_t)lds_dst),
                  "v"((unsigned long long)(uintptr_t)gsrc)
               : "memory");
}
__device__ __forceinline__ void async_wait0() {
  asm volatile("s_wait_asynccnt 0x0" ::: "memory");
}

// ---------------- Kernel 1: decode boxes (both models, flip model 2) --------
__global__ __launch_bounds__(256) void decode_kernel(
    const float* __restrict__ loc1, const float* __restrict__ loc2,
    const float* __restrict__ dbox, float* __restrict__ boxes) {
  int g = blockIdx.x * blockDim.x + threadIdx.x;
  if (g >= BB * NN) return;
  int b = g / NN;
  int n = g - b * NN;
  float4 d  = ((const float4*)dbox)[n];
  float4 l1 = ((const float4*)loc1)[g];
  float4 l2 = ((const float4*)loc2)[g];

  // model 1: cxcy = d + (l*0.1)*d ; wh = d * exp(l*0.2)
  float cx = d.x + (l1.x * 0.1f) * d.x;
  float cy = d.y + (l1.y * 0.1f) * d.y;
  float w  = d.z * expf(l1.z * 0.2f);
  float h  = d.w * expf(l1.w * 0.2f);
  float x1 = cx - w * 0.5f;
  float y1 = cy - h * 0.5f;
  float4 o1 = make_float4(x1, y1, x1 + w, y1 + h);
  ((float4*)boxes)[(size_t)b * N2 + n] = o1;

  // model 2 + horizontal flip: [1-x2, y1, 1-x1, y2]
  cx = d.x + (l2.x * 0.1f) * d.x;
  cy = d.y + (l2.y * 0.1f) * d.y;
  w  = d.z * expf(l2.z * 0.2f);
  h  = d.w * expf(l2.w * 0.2f);
  x1 = cx - w * 0.5f;
  y1 = cy - h * 0.5f;
  float4 o2 = make_float4(1.0f - (x1 + w), y1, 1.0f - x1, y1 + h);
  ((float4*)boxes)[(size_t)b * N2 + NN + n] = o2;
}

// ---------------- Kernel 2: softmax -> class-major scores (B, C, 2N) --------
__global__ __launch_bounds__(256) void softmax_kernel(
    const float* __restrict__ conf1, const float* __restrict__ conf2,
    float* __restrict__ scores) {
  int g = blockIdx.x * blockDim.x + threadIdx.x;
  if (g >= BB * NN) return;
  int b = g / NN;
  int n = g - b * NN;
  const float* c1 = conf1 + (size_t)g * CC;
  const float* c2 = conf2 + (size_t)g * CC;
  size_t base = ((size_t)b * CC) * (size_t)N2 + (size_t)n;

  float x[CC];
  // model 1
  float m = -INFINITY;
#pragma unroll
  for (int c = 0; c < CC; ++c) { x[c] = c1[c]; m = fmaxf(m, x[c]); }
  float s = 0.f;
#pragma unroll
  for (int c = 0; c < CC; ++c) { x[c] = expf(x[c] - m); s += x[c]; }
#pragma unroll
  for (int c = 0; c < CC; ++c) scores[base + (size_t)c * N2] = x[c] / s;

  // model 2
  m = -INFINITY;
#pragma unroll
  for (int c = 0; c < CC; ++c) { x[c] = c2[c]; m = fmaxf(m, x[c]); }
  s = 0.f;
#pragma unroll
  for (int c = 0; c < CC; ++c) { x[c] = expf(x[c] - m); s += x[c]; }
#pragma unroll
  for (int c = 0; c < CC; ++c) scores[base + (size_t)c * N2 + NN] = x[c] / s;
}

// ---------------- Kernel 3: per-(b,c) LDS top-k + wave32 NMS + output -------
__global__ __launch_bounds__(256) void topk_nms_kernel(
    const float* __restrict__ scores, const float* __restrict__ boxes,
    float* __restrict__ out) {
  // CDNA5: 320KB LDS/WGP lets us stage the full 17464-entry score vector.
  __shared__ uint32_t s_keys[N2];               // 69856 B (floats, then keys)
  __shared__ unsigned long long s_red[8];
  __shared__ unsigned long long s_win;
  __shared__ uint32_t s_topkey[TOPK];
  __shared__ int s_topidx[TOPK];
  __shared__ float s_tb[TOPK * 4];
  __shared__ int s_keepf[TOPK];
  __shared__ int s_map[TOPK];
  __shared__ int s_flag;

  const int tid = threadIdx.x;
  const int bc = blockIdx.x;
  const int b = bc / CC;
  const int c = bc - b * CC;
  const float* sc = scores + (size_t)bc * (size_t)N2;
  const uint32_t KEY_TH = fkey(CONF_TH);

  if (tid == 0) s_flag = 0;

  // --- async-stage raw scores into LDS (16B per lane per op) ---
  for (int e4 = tid; e4 < N2V4; e4 += 256)
    async_copy_b128(&s_keys[e4 * 4], sc + (size_t)e4 * 4);
  async_wait0();
  __syncthreads();

  // in-place transform: float -> masked monotonic key; any(s1>thresh) flag
  for (int i = tid; i < N2; i += 256) {
    float v = __uint_as_float(s_keys[i]);
    if (i < NN && v > CONF_TH) s_flag = 1;
    float mv = (v > CONF_TH) ? v : -1.0f;
    s_keys[i] = fkey(mv);
  }
  __syncthreads();

  // per-thread chunk top-2 cache (best + second-best)
  const int lo = tid * CHUNK;
  int hi = lo + CHUNK; if (hi > N2) hi = N2;
  unsigned long long best = 0ull, snd = 0ull;
  for (int i = lo; i < hi; ++i) {
    unsigned long long k = ((unsigned long long)s_keys[i] << 32) |
                           (uint32_t)(0xFFFFFFFFu - (uint32_t)i);
    if (k > best) { snd = best; best = k; }
    else if (k > snd) { snd = k; }
  }
  bool have_snd = true;

  const int lane = tid & 31;
  const int wid  = tid >> 5;

  // exact top-200 (desc value, asc index on ties): tournament w/ cached chunks
  for (int pass = 0; pass < TOPK; ++pass) {
    unsigned long long v = best;
#pragma unroll
    for (int off = 16; off > 0; off >>= 1) {
      unsigned long long o = __shfl_xor(v, off, 32);
      if (o > v) v = o;
    }
    if (lane == 0) s_red[wid] = v;
    __syncthreads();
    if (tid == 0) {
      unsigned long long w = s_red[0];
#pragma unroll
      for (int j = 1; j < 8; ++j) if (s_red[j] > w) w = s_red[j];
      s_win = w;
      s_topkey[pass] = (uint32_t)(w >> 32);
      s_topidx[pass] = (int)(0xFFFFFFFFu - (uint32_t)(w & 0xFFFFFFFFull));
    }
    __syncthreads();
    int widx = (int)(0xFFFFFFFFu - (uint32_t)(s_win & 0xFFFFFFFFull));
    if (widx >= lo && widx < hi) {     // winner is my chunk's current best
      s_keys[widx] = 0;
      if (have_snd) { best = snd; have_snd = false; }  // O(1) promote
      else {                                            // rescan -> top-2
        best = 0ull; snd = 0ull;
        for (int i = lo; i < hi; ++i) {
          unsigned long long k = ((unsigned long long)s_keys[i] << 32) |
                                 (uint32_t)(0xFFFFFFFFu - (uint32_t)i);
          if (k > best) { snd = best; best = k; }
          else if (k > snd) { snd = k; }
        }
        have_snd = true;
      }
    }
  }

  // async-gather top-200 boxes into LDS (random-index 16B copies)
  for (int r = tid; r < TOPK; r += 256) {
    int idx = s_topidx[r];
    async_copy_b128(&s_tb[r * 4], boxes + ((size_t)b * N2 + idx) * 4);
  }
  async_wait0();
  __syncthreads();

  // sequential NMS on wave 0 only: lane owns rows lane+32t, keep bits in regs,
  // wave32 ballot (__any) replaces barriers entirely.
  if (wid == 0) {
    uint32_t keep = 0;
    for (int i = 0; i < TOPK; ++i) {
      float bx0 = s_tb[i * 4 + 0], by0 = s_tb[i * 4 + 1];
      float bx1 = s_tb[i * 4 + 2], by1 = s_tb[i * 4 + 3];
      float areaB = (bx1 - bx0) * (by1 - by0);
      bool sup = false;
#pragma unroll
      for (int t = 0; t < 7; ++t) {
        int r = lane + 32 * t;
        if (r < TOPK && ((keep >> t) & 1u)) {
          float a0 = s_tb[r * 4 + 0], a1 = s_tb[r * 4 + 1];
          float a2 = s_tb[r * 4 + 2], a3 = s_tb[r * 4 + 3];
          float xx1 = fmaxf(a0, bx0), yy1 = fmaxf(a1, by0);
          float xx2 = fminf(a2, bx1), yy2 = fminf(a3, by1);
          float inter = fmaxf(xx2 - xx1, 0.f) * fmaxf(yy2 - yy1, 0.f);
          float areaA = (a2 - a0) * (a3 - a1);
          float iou = inter / (areaA + areaB - inter);
          if (iou > NMS_TH) sup = true;
        }
      }
      bool anysup = __any(sup);
      if (lane == (i & 31)) {
        bool tv = s_topkey[i] > KEY_TH;
        if (tv && !anysup) keep |= (1u << (i >> 5));
      }
    }
#pragma unroll
    for (int t = 0; t < 7; ++t) {
      int r = lane + 32 * t;
      if (r < TOPK) s_keepf[r] = (int)((keep >> t) & 1u);
    }
  }
  __syncthreads();

  // compaction map: output row r -> source top row, or -1
  if (tid == 0) {
    int rank = 0;
    for (int i = 0; i < TOPK; ++i)
      if (s_keepf[i]) s_map[rank++] = i;
    for (int r = rank; r < TOPK; ++r) s_map[r] = -1;
  }
  __syncthreads();

  const bool live = (c != 0) && (s_flag != 0);  // class 0 zeroed; any(s1>th) gate
  float* op = out + (size_t)bc * (size_t)(TOPK * 5);
  for (int r = tid; r < TOPK; r += 256) {
    int si = live ? s_map[r] : -1;
    float v0 = 0.f, v1 = 0.f, v2 = 0.f, v3 = 0.f, v4 = 0.f;
    if (si >= 0) {
      v0 = unkey(s_topkey[si]);
      v1 = s_tb[si * 4 + 0]; v2 = s_tb[si * 4 + 1];
      v3 = s_tb[si * 4 + 2]; v4 = s_tb[si * 4 + 3];
    }
    op[r * 5 + 0] = v0; op[r * 5 + 1] = v1; op[r * 5 + 2] = v2;
    op[r * 5 + 3] = v3; op[r * 5 + 4] = v4;
  }
}

// ---------------------------------------------------------------------------
extern "C" void kernel_launch(void* const* d_in, const int* in_sizes, int n_in,
                              void* d_out, int out_size, void* d_ws, size_t ws_size,
                              hipStream_t stream) {
  const float* loc1  = (const float*)d_in[0];
  const float* conf1 = (const float*)d_in[1];
  const float* loc2  = (const float*)d_in[2];
  const float* conf2 = (const float*)d_in[3];
  const float* dbox  = (const float*)d_in[4];
  float* out = (float*)d_out;

  // workspace layout: boxes (B,2N,4) f32 ; scores (B,C,2N) f32  (~112 MB)
  float* d_boxes  = (float*)d_ws;
  float* d_scores = d_boxes + (size_t)BB * N2 * 4;

  const int nthreads = BB * NN;
  const int blocks = (nthreads + 255) / 256;

  decode_kernel<<<blocks, 256, 0, stream>>>(loc1, loc2, dbox, d_boxes);
  softmax_kernel<<<blocks, 256, 0, stream>>>(conf1, conf2, d_scores);
  topk_nms_kernel<<<BB * CC, 256, 0, stream>>>(d_scores, d_boxes, out);
}